// MosaicConvModule_49460843381047
// MI455X (gfx1250) — compile-verified
//
#include <hip/hip_runtime.h>

typedef float v2f __attribute__((ext_vector_type(2)));
typedef float v4f __attribute__((ext_vector_type(4)));
typedef float v8f __attribute__((ext_vector_type(8)));

#define MSFA  4
#define KS    3
#define META  128
#define OUTC  16
#define QDIM  (KS * KS * OUTC)   // 144
#define IMG_H 1024
#define IMG_W 1024
#define IMG_B 8

// ---------------------------------------------------------------------------
// Kernel 1: Pos2Weight MLP (16 phases x relu(pos@W1+b1)@W2+b2), then swizzle
// the 9x16 per-phase weights into CDNA5 WMMA f32 16x16x4 B-fragment layout:
//   B 4x16: VGPR v, lane l -> K = v + 2*(l/16), N = l%16
// K is padded 9 -> 12 (three K=4 chunks), pad rows are zero.
// Workspace layout: wsB[ph][chunk][vgpr][lane], 16*3*2*32 floats = 12 KB.
// ---------------------------------------------------------------------------
__global__ void mosaic_mlp_kernel(const float* __restrict__ W1,
                                  const float* __restrict__ b1,
                                  const float* __restrict__ W2,
                                  const float* __restrict__ b2,
                                  float* __restrict__ wsB) {
  __shared__ float hidden[META];
  __shared__ float wq[QDIM];
  const int ph = blockIdx.x;          // 0..15, ph = i*4 + j
  const int i = ph >> 2, j = ph & 3;
  const float p0 = (i + 1) * 0.25f;   // h offset
  const float p1 = (j + 1) * 0.25f;   // w offset
  const int t = threadIdx.x;

  if (t < META) {
    float h = p0 * W1[t] + p1 * W1[META + t] + b1[t];
    hidden[t] = h > 0.0f ? h : 0.0f;
  }
  __syncthreads();
  if (t < QDIM) {
    float acc = b2[t];
    for (int m = 0; m < META; ++m) acc += hidden[m] * W2[m * QDIM + t];
    wq[t] = acc;                      // wq[p*16 + c]
  }
  __syncthreads();
  if (t < 192) {                      // 3 chunks * 2 VGPRs * 32 lanes
    int kk  = t / 64;
    int rem = t % 64;
    int v   = rem / 32;
    int l   = rem % 32;
    int k   = kk * 4 + v + 2 * (l / 16);     // padded K index 0..11
    int n   = l % 16;                        // output channel
    float val = (k < 9) ? wq[k * OUTC + n] : 0.0f;
    wsB[((ph * 3 + kk) * 2 + v) * 32 + l] = val;
  }
}

// ---------------------------------------------------------------------------
// Kernel 2: main mosaic conv. One block = 8 waves = 8 consecutive rows x 64
// columns of one batch image. Each wave runs 4 phase-tiles (j = x%4), each
// tile = 3x V_WMMA_F32_16X16X4_F32 (M=16 pixels, N=16 channels, K=12).
// D-fragments are interleaved through LDS so global stores are contiguous
// float4 runs per channel row (store-BW bound kernel -> full sector use).
// ---------------------------------------------------------------------------
__global__ __launch_bounds__(256)
void mosaic_conv_kernel(const float* __restrict__ in,
                        const float* __restrict__ wsB,
                        float* __restrict__ out) {
  __shared__ __align__(16) float tile[10 * 68];    // rows y0-1..y0+8, cols x0-1..x0+64
  __shared__ __align__(16) float stage[8 * 1024];  // per-wave [16 ch][64 x]

  const int x0  = blockIdx.x * 64;
  const int y0  = blockIdx.y * 8;
  const int b   = blockIdx.z;
  const int tid = threadIdx.x;

  // Cooperative load of the 10x66 zero-padded input window.
  for (int idx = tid; idx < 10 * 66; idx += 256) {
    int r = idx / 66, c = idx % 66;
    int gy = y0 - 1 + r, gx = x0 - 1 + c;
    float v = 0.0f;
    if ((unsigned)gy < IMG_H && (unsigned)gx < IMG_W)
      v = in[((size_t)b * IMG_H + gy) * IMG_W + gx];
    tile[r * 68 + c] = v;
  }
  __syncthreads();

  const int wave  = tid >> 5;     // row within the stripe (y = y0 + wave)
  const int lane  = tid & 31;
  const int y     = y0 + wave;
  const int iph   = y & 3;        // mosaic row phase
  const int thalf = lane >> 4;    // half-wave selector (A/B K striping)
  const int tcol  = lane & 15;

  float* st = stage + wave * 1024;

  for (int jph = 0; jph < 4; ++jph) {
    v8f d = {};
    const float* bbase = wsB + (size_t)((iph * 4 + jph) * 3) * 64;
#pragma unroll
    for (int kk = 0; kk < 3; ++kk) {
      v2f a, bm;
#pragma unroll
      for (int v = 0; v < 2; ++v) {
        int k = kk * 4 + v + 2 * thalf;          // padded patch index 0..11
        float av = 0.0f;
        if (k < 9) {
          int dy = k / 3, dx = k % 3;            // kernel offset (row-major, pad=1)
          av = tile[(wave + dy) * 68 + (jph + 4 * tcol + dx)];
        }
        a[v]  = av;
        bm[v] = bbase[(kk * 2 + v) * 32 + lane]; // pre-swizzled B fragment
      }
      // D(16x16 f32) += A(16x4 f32) x B(4x16 f32)
      d = __builtin_amdgcn_wmma_f32_16x16x4_f32(false, a, false, bm,
                                                (short)0, d, false, false);
    }
    // Scatter D fragment: VGPR r, lane l -> pixel M = r + 8*(l/16), chan N = l%16
#pragma unroll
    for (int r = 0; r < 8; ++r) {
      int m = r + 8 * thalf;          // pixel index 0..15
      int c = tcol;                   // channel 0..15
      st[c * 64 + (jph + 4 * m)] = d[r];
    }
  }

  // Coalesced write-out: lane pair (2 lanes) covers one channel's 64 pixels.
  const int c    = lane >> 1;
  const int half = lane & 1;
  float* obase = out + (((size_t)b * OUTC + c) * IMG_H + y) * IMG_W + x0 + half * 32;
#pragma unroll
  for (int g = 0; g < 8; ++g) {
    v4f vv = *(const v4f*)(st + c * 64 + half * 32 + g * 4);
    __builtin_nontemporal_store(vv, (v4f*)(obase + g * 4));
  }
}

// ---------------------------------------------------------------------------
extern "C" void kernel_launch(void* const* d_in, const int* in_sizes, int n_in,
                              void* d_out, int out_size, void* d_ws, size_t ws_size,
                              hipStream_t stream) {
  const float* input = (const float*)d_in[0];
  const float* W1    = (const float*)d_in[1];
  const float* b1    = (const float*)d_in[2];
  const float* W2    = (const float*)d_in[3];
  const float* b2    = (const float*)d_in[4];
  float* out = (float*)d_out;
  float* wsB = (float*)d_ws;   // 12 KB of B-fragments

  mosaic_mlp_kernel<<<dim3(16), dim3(192), 0, stream>>>(W1, b1, W2, b2, wsB);
  mosaic_conv_kernel<<<dim3(IMG_W / 64, IMG_H / 8, IMG_B), dim3(256), 0, stream>>>(
      input, wsB, out);
}